// FusedSparseModules_18597208392061
// MI455X (gfx1250) — compile-verified
//
#include <hip/hip_runtime.h>
#include <hip/hip_bf16.h>

// EmbeddingBag(mode='sum') fused gather + segment-sum for MI455X (gfx1250).
//   values : int32 [V]                 (V = 2,097,152)
//   offsets: int32 [NUM_BAGS+1]        (bags feature-major: bag = f*B + b)
//   weight : float32 [VOCAB, 64]
//   out    : float32 [B, F, 64]  (= pooled.view(F,B,D).transpose(0,1))
//
// Memory-bound gather-reduce (~0.25 FLOP/byte). One wave32 per bag:
//  * bag's contiguous index slice staged into LDS via
//    global_load_async_to_lds_b128 (ASYNCcnt + s_wait_asynccnt),
//  * 2 embedding rows per iteration: 16 lanes x float4 = one 256B row,
//  * global_prefetch_b8 for rows ~8 ahead (index sourced from LDS),
//  * cross-half reduce via wave32 __shfl_xor(16), coalesced 256B store.

#define EB_F 26
#define EB_B 4096
#define EB_D 64
#define EB_NUM_BAGS (EB_F * EB_B)

#define EB_WAVES_PER_BLOCK 8
#define EB_TILE_IDX 128      // LDS index buffer entries per wave
#define EB_TILE_STRIDE 124   // rows consumed per tile (covers 16B-align skew)

__global__ __launch_bounds__(EB_WAVES_PER_BLOCK * 32) void
embbag_sum_kernel(const int* __restrict__ values,
                  const int* __restrict__ offsets,
                  const float* __restrict__ weight,
                  float* __restrict__ out,
                  int vtot)
{
    __shared__ int sIdx[EB_WAVES_PER_BLOCK][EB_TILE_IDX];

    const int tid  = blockIdx.x * blockDim.x + threadIdx.x;
    const int bag  = tid >> 5;                       // one wave32 per bag
    const int lane = threadIdx.x & 31;
    const int w    = (threadIdx.x >> 5) & (EB_WAVES_PER_BLOCK - 1);
    if (bag >= EB_NUM_BAGS) return;

    const int start = offsets[bag];
    const int end   = offsets[bag + 1];

    const int half = lane >> 4;          // 0: even rows, 1: odd rows
    const int sub  = lane & 15;          // 16-byte chunk within a 256-byte row

    float4 acc = make_float4(0.f, 0.f, 0.f, 0.f);

    const int PF = 8;                    // prefetch distance in rows

    for (int tileStart = start; tileStart < end; tileStart += EB_TILE_STRIDE) {
        const int tileEnd = min(tileStart + EB_TILE_STRIDE, end);
        const int abase   = tileStart & ~3;   // 16B-aligned copy origin

        // WAR fence: prior ds reads of this buffer must land before the
        // async engine overwrites it (async LDS writes are unordered vs DS).
        asm volatile("s_wait_dscnt 0x0" ::: "memory");

        // Stage 128 indices: 32 lanes x 16B = 512B, one async instruction.
        {
            int co = abase + lane * 4;
            if (co > vtot - 4) co = vtot - 4;          // stay in-bounds
            unsigned long long ga =
                (unsigned long long)(uintptr_t)(values + co);
            unsigned ldsOff =
                (unsigned)(uintptr_t)(&sIdx[w][lane * 4]); // addr[31:0] = LDS off
            asm volatile("global_load_async_to_lds_b128 %0, %1, off"
                         :: "v"(ldsOff), "v"(ga) : "memory");
            asm volatile("s_wait_asynccnt 0x0" ::: "memory");
        }

        #pragma unroll 4
        for (int i = tileStart; i < tileEnd; i += 2) {
            const int rl = i - abase;      // local slot of row i in sIdx

            // --- prefetch a future row's cachelines (global_prefetch_b8) ---
            if (i + PF + half < tileEnd) {
                const int pidx = sIdx[w][rl + PF + half];
                __builtin_prefetch(weight + (size_t)pidx * EB_D + sub * 4, 0, 0);
            }

            // --- gather + accumulate: 2 rows per wave iteration ---
            const int row = i + half;
            if (row < tileEnd) {
                const int idx = sIdx[w][rl + half];     // LDS broadcast
                const float4* __restrict__ p =
                    (const float4*)(weight + (size_t)idx * EB_D);
                const float4 v = p[sub];   // global_load_b128: 16 lanes = 1 row
                acc.x += v.x;
                acc.y += v.y;
                acc.z += v.z;
                acc.w += v.w;
            }
        }
    }

    // cross-half reduction: lane l += lane (l^16)   (wave32 shuffle)
    acc.x += __shfl_xor(acc.x, 16);
    acc.y += __shfl_xor(acc.y, 16);
    acc.z += __shfl_xor(acc.z, 16);
    acc.w += __shfl_xor(acc.w, 16);

    if (half == 0) {
        // bag = f*B + b  ->  out[b][f][:]
        const int f = bag / EB_B;
        const int b = bag - f * EB_B;
        float4* __restrict__ o =
            (float4*)(out + ((size_t)b * EB_F + f) * EB_D);
        o[sub] = acc;                      // 16 lanes x 16B = 256B coalesced
    }
}

extern "C" void kernel_launch(void* const* d_in, const int* in_sizes, int n_in,
                              void* d_out, int out_size, void* d_ws, size_t ws_size,
                              hipStream_t stream)
{
    (void)n_in; (void)d_ws; (void)ws_size; (void)out_size;

    const int*   values  = (const int*)d_in[0];
    const int*   offsets = (const int*)d_in[1];
    const float* weight  = (const float*)d_in[2];
    float*       out     = (float*)d_out;
    const int    vtot    = in_sizes[0];

    const int threads = EB_WAVES_PER_BLOCK * 32;
    const int blocks  = (EB_NUM_BAGS + EB_WAVES_PER_BLOCK - 1) / EB_WAVES_PER_BLOCK;

    embbag_sum_kernel<<<blocks, threads, 0, stream>>>(values, offsets, weight,
                                                      out, vtot);
}